// GCN_31344671326396
// MI455X (gfx1250) — compile-verified
//
#include <hip/hip_runtime.h>
#include <hip/hip_bf16.h>

// CDNA5 wave32 WMMA fragments
typedef float v2f __attribute__((ext_vector_type(2)));
typedef float v8f __attribute__((ext_vector_type(8)));

#define DIN 128  // input feature width for every layer's GEMM (D_IN == D_HID == 128)

// ---------------------------------------------------------------------------
// Small utility kernels
// ---------------------------------------------------------------------------
__global__ void zero_f32(float* __restrict__ p, int n) {
    int i = blockIdx.x * blockDim.x + threadIdx.x;
    if (i < n) p[i] = 0.0f;
}

__global__ void deg_accum(const int* __restrict__ dst, const float* __restrict__ ew,
                          float* __restrict__ deg, int E) {
    int e = blockIdx.x * blockDim.x + threadIdx.x;
    if (e < E) atomicAdd(deg + dst[e], ew[e]);
}

__global__ void deg_to_dinv(float* __restrict__ deg, int n) {
    int i = blockIdx.x * blockDim.x + threadIdx.x;
    if (i < n) {
        float d = deg[i];
        deg[i] = (d > 0.0f) ? rsqrtf(d) : 0.0f;  // in-place: deg -> deg^{-1/2}
    }
}

template <int D>
__global__ void bias_init(float* __restrict__ out, const float* __restrict__ b, int n) {
    long long i = (long long)blockIdx.x * blockDim.x + threadIdx.x;
    if (i < (long long)n * D) out[i] = b[(int)(i % D)];
}

__global__ void relu_ip(float* __restrict__ p, long long n) {
    long long i = (long long)blockIdx.x * blockDim.x + threadIdx.x;
    if (i < n) {
        float v = p[i];
        p[i] = (v > 0.0f) ? v : 0.0f;
    }
}

// ---------------------------------------------------------------------------
// Dense h = X @ W via V_WMMA_F32_16X16X4_F32 (exact f32, matrix pipe).
// One block per 16-row tile of X; wave w computes the 16x16 output tile for
// columns [16w, 16w+16). K is looped in steps of 4 (DIN/4 = 32 WMMAs).
//
// Fragment layouts per CDNA5 ISA 7.12.2 (wave32):
//   A 16x4 f32 : lanes 0-15 -> rows 0-15; VGPR v holds K = kb + 2*half + v
//   B 4x16 f32 : symmetric (lane -> col, VGPR v -> K = kb + 2*half + v)
//   C/D 16x16  : lane r -> col r; VGPR v -> row v + 8*half
// ---------------------------------------------------------------------------
template <int DOUT>
__global__ __launch_bounds__(32 * (DOUT / 16))
void gemm_xw_wmma(const float* __restrict__ X, const float* __restrict__ W,
                  float* __restrict__ H, int n) {
    // +4 pad: LDS row stride 132 floats -> the 16 lanes reading 16 different
    // rows at the same K hit banks (4*r + c) % 64, all distinct.
    __shared__ float As[16][DIN + 4];

    const int rowBase = blockIdx.x * 16;
    const int tid     = threadIdx.x;
    const int nthr    = 32 * (DOUT / 16);

    // Cooperative load of the 16 x 128 A tile (zero-pad past n).
    for (int i = tid; i < 16 * DIN; i += nthr) {
        int r = i / DIN, c = i % DIN;
        int row = rowBase + r;
        As[r][c] = (row < n) ? X[(size_t)row * DIN + c] : 0.0f;
    }
    __syncthreads();

    const int wave = tid >> 5;
    const int lane = tid & 31;
    const int half = lane >> 4;   // 0: lanes 0-15, 1: lanes 16-31
    const int r    = lane & 15;
    const int col  = wave * 16 + r;

    v8f acc = {};
#pragma unroll
    for (int kb = 0; kb < DIN; kb += 4) {
        const int k0 = kb + 2 * half;
        v2f a, b;
        a.x = As[r][k0];
        a.y = As[r][k0 + 1];
        b.x = W[(size_t)k0 * DOUT + col];
        b.y = W[(size_t)(k0 + 1) * DOUT + col];
        acc = __builtin_amdgcn_wmma_f32_16x16x4_f32(
            /*neg_a=*/false, a, /*neg_b=*/false, b,
            /*c_mod=*/(short)0, acc, /*reuse_a=*/false, /*reuse_b=*/false);
    }

#pragma unroll
    for (int v = 0; v < 8; ++v) {
        int row = rowBase + v + 8 * half;
        if (row < n) H[(size_t)row * DOUT + col] = acc[v];
    }
}

// ---------------------------------------------------------------------------
// Edge aggregation: out[dst] += dinv[src]*ew*dinv[dst] * h[src]
// One wave covers one edge for D=128 (32 lanes x float4): coalesced gather of
// h[src] and 4 contiguous f32 atomics per lane into out[dst]. Both tables are
// L2-resident (51.2 MB < 192 MB L2) so this runs at L2 bandwidth.
// ---------------------------------------------------------------------------
template <int D>
__global__ void edge_aggregate(const float* __restrict__ H, const int* __restrict__ src,
                               const int* __restrict__ dst, const float* __restrict__ ew,
                               const float* __restrict__ dinv, float* __restrict__ out,
                               int E) {
    constexpr int CH = D / 4;  // float4 chunks per edge
    long long idx = (long long)blockIdx.x * blockDim.x + threadIdx.x;
    if (idx >= (long long)E * CH) return;
    int e = (int)(idx / CH);
    int c = (int)(idx % CH) * 4;

    int s = src[e];
    int d = dst[e];
    float norm = dinv[s] * ew[e] * dinv[d];

    const float4 hv = *reinterpret_cast<const float4*>(H + (size_t)s * D + c);
    float* o = out + (size_t)d * D + c;
    atomicAdd(o + 0, norm * hv.x);
    atomicAdd(o + 1, norm * hv.y);
    atomicAdd(o + 2, norm * hv.z);
    atomicAdd(o + 3, norm * hv.w);
}

// ---------------------------------------------------------------------------
// Host-side orchestration
// ---------------------------------------------------------------------------
extern "C" void kernel_launch(void* const* d_in, const int* in_sizes, int n_in,
                              void* d_out, int out_size, void* d_ws, size_t ws_size,
                              hipStream_t stream) {
    const float* x  = (const float*)d_in[0];
    const int*   ei = (const int*)d_in[1];   // [2, E] flat (harness: integer -> int*)
    const float* ew = (const float*)d_in[2];
    const float* W1 = (const float*)d_in[3];
    const float* b1 = (const float*)d_in[4];
    const float* W2 = (const float*)d_in[5];
    const float* b2 = (const float*)d_in[6];
    const float* W3 = (const float*)d_in[7];
    const float* b3 = (const float*)d_in[8];

    const int N = in_sizes[0] / DIN;
    const int E = in_sizes[1] / 2;
    const int* src = ei;
    const int* dst = ei + E;

    // Workspace: h scratch (N*128 f32) then deg/dinv (N f32) ~= 51.6 MB
    float* h    = (float*)d_ws;
    float* dinv = h + (size_t)N * DIN;

    // Output regions, concatenated in reference return order
    float* emb1 = (float*)d_out;                  // [N, 128]
    float* emb2 = emb1 + (size_t)N * 128;         // [N, 128]
    float* emb3 = emb2 + (size_t)N * 128;         // [N, 64]

    const int T = 256;
    dim3 gemmGrid((N + 15) / 16);

    // ---- gcn_norm: deg = scatter_add(ew, dst); dinv = deg^{-1/2} ----
    zero_f32<<<(N + T - 1) / T, T, 0, stream>>>(dinv, N);
    deg_accum<<<(E + T - 1) / T, T, 0, stream>>>(dst, ew, dinv, E);
    deg_to_dinv<<<(N + T - 1) / T, T, 0, stream>>>(dinv, N);

    long long nw128 = (long long)E * 32;  // threads for D=128 aggregation
    long long nw64  = (long long)E * 16;  // threads for D=64 aggregation

    // ---- Layer 1: emb1 = relu(agg(x @ W1) + b1) ----
    gemm_xw_wmma<128><<<gemmGrid, 256, 0, stream>>>(x, W1, h, N);
    bias_init<128><<<(int)(((long long)N * 128 + T - 1) / T), T, 0, stream>>>(emb1, b1, N);
    edge_aggregate<128><<<(int)((nw128 + T - 1) / T), T, 0, stream>>>(h, src, dst, ew, dinv, emb1, E);
    relu_ip<<<(int)(((long long)N * 128 + T - 1) / T), T, 0, stream>>>(emb1, (long long)N * 128);

    // ---- Layer 2: emb2 = relu(agg(emb1 @ W2) + b2) ----
    gemm_xw_wmma<128><<<gemmGrid, 256, 0, stream>>>(emb1, W2, h, N);
    bias_init<128><<<(int)(((long long)N * 128 + T - 1) / T), T, 0, stream>>>(emb2, b2, N);
    edge_aggregate<128><<<(int)((nw128 + T - 1) / T), T, 0, stream>>>(h, src, dst, ew, dinv, emb2, E);
    relu_ip<<<(int)(((long long)N * 128 + T - 1) / T), T, 0, stream>>>(emb2, (long long)N * 128);

    // ---- Layer 3: emb3 = agg(emb2 @ W3) + b3 (no relu) ----
    gemm_xw_wmma<64><<<gemmGrid, 128, 0, stream>>>(emb2, W3, h, N);
    bias_init<64><<<(int)(((long long)N * 64 + T - 1) / T), T, 0, stream>>>(emb3, b3, N);
    edge_aggregate<64><<<(int)((nw64 + T - 1) / T), T, 0, stream>>>(h, src, dst, ew, dinv, emb3, E);
}